// FNO2d_17927193494229
// MI455X (gfx1250) — compile-verified
//
// FNO2d for MI455X (gfx1250, wave32, WMMA). Compile-only target.
// All Fourier stages are truncated-DFT GEMMs on V_WMMA_F32_16X16X4_F32 with
// precomputed basis tables (zero-padded so no guarded loads are needed).
// Weight matrices for the pointwise conv and fc1 are staged through LDS.
// Required workspace: ~566 MB.
#include <hip/hip_runtime.h>
#include <math.h>

typedef float v2f __attribute__((ext_vector_type(2)));
typedef float v8f __attribute__((ext_vector_type(8)));

#define B_   32
#define S_   512
#define TI   10
#define TO   40
#define WID  64
#define XP   521            // 512 + PAD
#define TP   49             // 40 + PAD
#define NP   (XP * TP)      // 25529 spatial points (padded grid)
#define XPA  528            // X rows padded to multiple of 16 (DFT buffers)
#define TPK  52             // t K-dim padded to multiple of 4
#define XPK  524            // x K-dim padded to multiple of 4
#define XT_TILES 33         // ceil(521/16)
#define PT_TILES 1596       // ceil(25529/16)
#define SLACK 512           // tail slack (floats) for unguarded K-tail reads
#define PI2  6.28318530717958647692f

// --------------------------------------------------------------------------
// wave32 f32 WMMA: D(16x16) = A(16x4) * B(4x16) + C
// Fragment layouts (CDNA5 ISA 7.12.2):
//   A: lane l -> row m = l&15 ; elem0 k = 2*(l>>4), elem1 k = 2*(l>>4)+1
//   B: lane l -> col n = l&15 ; same k mapping
//   D: vgpr v, lane l -> m = v + 8*(l>>4), n = l&15
// --------------------------------------------------------------------------
__device__ __forceinline__ v8f wmma4(v2f a, v2f b, v8f c) {
  return __builtin_amdgcn_wmma_f32_16x16x4_f32(
      /*neg_a=*/false, a, /*neg_b=*/false, b,
      /*c_mod=*/(short)0, c, /*reuse_a=*/false, /*reuse_b=*/false);
}

__device__ __forceinline__ float gelu_f(float x) {
  return 0.5f * x * (1.0f + erff(x * 0.70710678118654752440f));
}

// ===========================================================================
// Basis tables (computed once; ~80 KB total, L2-resident):
//  tEtR/tEtI [TPK][16] : forward-t  exp(-2pi i kt t/49); rows t>=49 zero
//  tCx /tSx  [XPA][16] : cos/sin(2pi x kx/521);          rows x>=521 zero
//  tBtR/tBtI [16][64]  : inverse-t real basis with Hermitian fold and
//                        1/(521*49) scale baked in;      cols t>=49 zero
// ===========================================================================
__global__ void k_tables(float* __restrict__ tEtR, float* __restrict__ tEtI,
                         float* __restrict__ tCx,  float* __restrict__ tSx,
                         float* __restrict__ tBtR, float* __restrict__ tBtI) {
  int i = blockIdx.x * blockDim.x + threadIdx.x;
  if (i < TPK * 16) {
    int t = i >> 4, kt = i & 15;
    float s, c;
    __sincosf(PI2 / (float)TP * (float)(kt * t), &s, &c);
    bool v = (t < TP);
    tEtR[i] = v ? c : 0.f;
    tEtI[i] = v ? -s : 0.f;
  }
  if (i < XPA * 16) {
    int x = i >> 4, kx = i & 15;
    float s, c;
    __sincosf(PI2 / (float)XP * (float)(kx * x), &s, &c);
    bool v = (x < XP);
    tCx[i] = v ? c : 0.f;
    tSx[i] = v ? s : 0.f;
  }
  if (i < 16 * 64) {
    int kt = i >> 6, t = i & 63;
    float s, c;
    __sincosf(PI2 / (float)TP * (float)(kt * t), &s, &c);
    bool v = (t < TP);
    const float al = 1.0f / ((float)XP * (float)TP);
    tBtR[i] = v ? (kt == 0 ? al : 2.0f * al * c) : 0.f;
    tBtI[i] = (v && kt != 0) ? -2.0f * al * s : 0.f;
  }
}

// ===========================================================================
// fc0: build padded activation v[b][c][x][t], (B,64,521,49), zero in pad.
// K=14 is too small for WMMA to pay off; elementwise kernel.
// ===========================================================================
__global__ void k_fc0(const float* __restrict__ u, const float* __restrict__ xin,
                      const float* __restrict__ tin, const float* __restrict__ par,
                      const float* __restrict__ fw, const float* __restrict__ fb,
                      float* __restrict__ vout, long long total) {
  long long i = (long long)blockIdx.x * blockDim.x + threadIdx.x;
  if (i >= total) return;
  int t = (int)(i % TP);
  long long r = i / TP;
  int x = (int)(r % XP); r /= XP;
  int c = (int)(r % WID);
  int b = (int)(r / WID);
  float val = 0.f;
  if (x < S_ && t < TO) {
    val = fb[c];
    const float* ub = u + ((long long)b * S_ + x) * TI;
#pragma unroll
    for (int k = 0; k < TI; ++k) val += ub[k] * fw[k * WID + c];
    val += par[b * 2 + 0] * fw[10 * WID + c];
    val += par[b * 2 + 1] * fw[11 * WID + c];
    val += xin[b * S_ + x] * fw[12 * WID + c];
    val += tin[b * TO + t] * fw[13 * WID + c];
  }
  vout[i] = val;
}

// ===========================================================================
// Forward DFT over t: W1[b,c,x,kt] = sum_t v[x,t] * Et[t,kt]
// All loads unguarded: K-tail (t 49..51) killed by zero basis rows, M-tail
// (x 521..527) lands in padded W1 rows and is killed downstream by zero tCx.
// ===========================================================================
__global__ void k_dft_t(const float* __restrict__ v, const float* __restrict__ tEtR,
                        const float* __restrict__ tEtI, float* __restrict__ w1r,
                        float* __restrict__ w1i, int nwaves) {
  int wid = (int)((blockIdx.x * blockDim.x + threadIdx.x) >> 5);
  if (wid >= nwaves) return;
  int lane = threadIdx.x & 31, half = lane >> 4, l16 = lane & 15;
  int xt = wid % XT_TILES;
  int bc = wid / XT_TILES;
  int x0 = xt * 16;
  const float* vb = v + (size_t)bc * NP;
  int xa = x0 + l16;                               // A row (x), may run past 520
  v8f cr = {}, ci = {};
  for (int kb = 0; kb < TPK; kb += 4) {
    int k0 = kb + 2 * half;
    v2f a, br, bi;
    a.x = vb[xa * TP + k0];
    a.y = vb[xa * TP + k0 + 1];
    br.x = tEtR[k0 * 16 + l16];  br.y = tEtR[(k0 + 1) * 16 + l16];
    bi.x = tEtI[k0 * 16 + l16];  bi.y = tEtI[(k0 + 1) * 16 + l16];
    cr = wmma4(a, br, cr);
    ci = wmma4(a, bi, ci);
  }
  float* wr = w1r + (size_t)bc * XPA * 16;
  float* wi = w1i + (size_t)bc * XPA * 16;
#pragma unroll
  for (int vv = 0; vv < 8; ++vv) {
    int x = x0 + vv + 8 * half;                    // < 528, always in bounds
    wr[x * 16 + l16] = cr[vv];
    wi[x * 16 + l16] = ci[vv];
  }
}

// ===========================================================================
// Forward DFT over x: vft[kx,kt] = sum_x Ex[kx,x] * W1[x,kt] (complex)
// A = (cos,-sin) from tables (zero rows kill x>=521 garbage in W1 pad rows).
// ===========================================================================
__global__ void k_dft_x(const float* __restrict__ w1r, const float* __restrict__ w1i,
                        const float* __restrict__ tCx, const float* __restrict__ tSx,
                        float* __restrict__ vftr, float* __restrict__ vfti, int nwaves) {
  int wid = (int)((blockIdx.x * blockDim.x + threadIdx.x) >> 5);
  if (wid >= nwaves) return;
  int lane = threadIdx.x & 31, half = lane >> 4, l16 = lane & 15;
  int bc = wid;
  const float* rr = w1r + (size_t)bc * XPA * 16;
  const float* ii = w1i + (size_t)bc * XPA * 16;
  v8f aR = {}, aI = {};
  for (int kb = 0; kb < XPK; kb += 4) {
    int x0c = kb + 2 * half, x1c = x0c + 1;        // <= 523 < 528
    v2f ac, as_, ans, br, bi;
    ac.x  = tCx[x0c * 16 + l16];  ac.y  = tCx[x1c * 16 + l16];  // A row = kx = l16
    as_.x = tSx[x0c * 16 + l16];  as_.y = tSx[x1c * 16 + l16];
    ans.x = -as_.x;               ans.y = -as_.y;
    br.x = rr[x0c * 16 + l16];    br.y = rr[x1c * 16 + l16];
    bi.x = ii[x0c * 16 + l16];    bi.y = ii[x1c * 16 + l16];
    // (cos - i sin)(Wr + i Wi): Re = cos*Wr + sin*Wi ; Im = cos*Wi - sin*Wr
    aR = wmma4(ac, br, aR);
    aR = wmma4(as_, bi, aR);
    aI = wmma4(ac, bi, aI);
    aI = wmma4(ans, br, aI);
  }
  float* orp = vftr + (size_t)bc * 256;
  float* oip = vfti + (size_t)bc * 256;
#pragma unroll
  for (int vv = 0; vv < 8; ++vv) {
    int m = vv + 8 * half;                         // kx
    orp[m * 16 + l16] = aR[vv];
    oip[m * 16 + l16] = aI[vv];
  }
}

// ===========================================================================
// Per-mode channel mix: corner[b,o,md] = sum_c vft[b,c,md] * w[c,o,md] (complex)
// 256 modes x (32x64)@(64x64) complex GEMM.
// ===========================================================================
__global__ void k_mix(const float* __restrict__ vftr, const float* __restrict__ vfti,
                      const float* __restrict__ wr, const float* __restrict__ wi,
                      float* __restrict__ corR, float* __restrict__ corI, int nwaves) {
  int wid = (int)((blockIdx.x * blockDim.x + threadIdx.x) >> 5);
  if (wid >= nwaves) return;
  int lane = threadIdx.x & 31, half = lane >> 4, l16 = lane & 15;
  int md = wid & 255;
  int rest = wid >> 8;
  int bt = rest & 1;                               // batch tile (2)
  int ot = rest >> 1;                              // out-channel tile (4)
  int bb = bt * 16 + l16;                          // A row = b
  int oo = ot * 16 + l16;                          // B col = o
  v8f accR = {}, accI = {};
  for (int kb = 0; kb < WID; kb += 4) {            // K = c
    int c0 = kb + 2 * half, c1 = c0 + 1;
    v2f ar, ai;
    ar.x = vftr[((size_t)(bb * WID + c0)) * 256 + md];
    ar.y = vftr[((size_t)(bb * WID + c1)) * 256 + md];
    ai.x = vfti[((size_t)(bb * WID + c0)) * 256 + md];
    ai.y = vfti[((size_t)(bb * WID + c1)) * 256 + md];
    v2f bR, bI, bIn;
    bR.x = wr[((size_t)(c0 * WID + oo)) * 256 + md];
    bR.y = wr[((size_t)(c1 * WID + oo)) * 256 + md];
    bI.x = wi[((size_t)(c0 * WID + oo)) * 256 + md];
    bI.y = wi[((size_t)(c1 * WID + oo)) * 256 + md];
    bIn.x = -bI.x; bIn.y = -bI.y;
    accR = wmma4(ar, bR, accR);
    accR = wmma4(ai, bIn, accR);
    accI = wmma4(ar, bI, accI);
    accI = wmma4(ai, bR, accI);
  }
#pragma unroll
  for (int vv = 0; vv < 8; ++vv) {
    int b = bt * 16 + vv + 8 * half;
    corR[((size_t)(b * WID + oo)) * 256 + md] = accR[vv];
    corI[((size_t)(b * WID + oo)) * 256 + md] = accI[vv];
  }
}

// ===========================================================================
// Inverse DFT over x: W2[x,kt] = sum_kx corner[kx,kt] * exp(+2pi i kx x/521)
// A from tables (zero rows => W2 pad rows 521..527 are exact zeros).
// ===========================================================================
__global__ void k_idft_x(const float* __restrict__ corR, const float* __restrict__ corI,
                         const float* __restrict__ tCx, const float* __restrict__ tSx,
                         float* __restrict__ w2r, float* __restrict__ w2i, int nwaves) {
  int wid = (int)((blockIdx.x * blockDim.x + threadIdx.x) >> 5);
  if (wid >= nwaves) return;
  int lane = threadIdx.x & 31, half = lane >> 4, l16 = lane & 15;
  int xt = wid % XT_TILES;
  int bo = wid / XT_TILES;
  int x0 = xt * 16;
  const float* cr = corR + (size_t)bo * 256;
  const float* ci = corI + (size_t)bo * 256;
  int xa = x0 + l16;                               // A row = x, < 528
  v8f Rr = {}, Ri = {};
  for (int kb = 0; kb < 16; kb += 4) {             // K = kx
    int k0 = kb + 2 * half, k1 = k0 + 1;
    v2f ac, as_, ans, bcr, bci;
    ac.x  = tCx[xa * 16 + k0];  ac.y  = tCx[xa * 16 + k1];
    as_.x = tSx[xa * 16 + k0];  as_.y = tSx[xa * 16 + k1];
    ans.x = -as_.x;             ans.y = -as_.y;
    bcr.x = cr[k0 * 16 + l16];  bcr.y = cr[k1 * 16 + l16];
    bci.x = ci[k0 * 16 + l16];  bci.y = ci[k1 * 16 + l16];
    // (cR + i cI)(cos + i sin): Re = cR cos - cI sin ; Im = cR sin + cI cos
    Rr = wmma4(ac, bcr, Rr);
    Rr = wmma4(ans, bci, Rr);
    Ri = wmma4(as_, bcr, Ri);
    Ri = wmma4(ac, bci, Ri);
  }
  float* wr = w2r + (size_t)bo * XPA * 16;
  float* wi = w2i + (size_t)bo * XPA * 16;
#pragma unroll
  for (int vv = 0; vv < 8; ++vv) {
    int x = x0 + vv + 8 * half;                    // < 528, in bounds
    wr[x * 16 + l16] = Rr[vv];
    wi[x * 16 + l16] = Ri[vv];
  }
}

// ===========================================================================
// Inverse real DFT over t: u1[x,t] = sum_kt W2r[x,kt]*BtR[kt,t] + W2i*BtI
// (Hermitian fold + 1/(X*T) scale are inside the table.) Writes u1 -> vnext.
// ===========================================================================
__global__ void k_idft_t(const float* __restrict__ w2r, const float* __restrict__ w2i,
                         const float* __restrict__ tBtR, const float* __restrict__ tBtI,
                         float* __restrict__ vnext, int nwaves) {
  int wid = (int)((blockIdx.x * blockDim.x + threadIdx.x) >> 5);
  if (wid >= nwaves) return;
  int lane = threadIdx.x & 31, half = lane >> 4, l16 = lane & 15;
  int tt = wid & 3;                                // 4 t-tiles cover 49
  int rest = wid >> 2;
  int xt = rest % XT_TILES;
  int bo = rest / XT_TILES;
  int x0 = xt * 16, t0 = tt * 16;
  const float* wr = w2r + (size_t)bo * XPA * 16;
  const float* wi = w2i + (size_t)bo * XPA * 16;
  int xa = x0 + l16;                               // A row = x, < 528 in bounds
  int ta = t0 + l16;                               // B col = t, < 64 (table padded)
  v8f acc = {};
  for (int kb = 0; kb < 16; kb += 4) {             // K = kt
    int k0 = kb + 2 * half, k1 = k0 + 1;
    v2f ar, ai, br, bi;
    ar.x = wr[xa * 16 + k0];  ar.y = wr[xa * 16 + k1];
    ai.x = wi[xa * 16 + k0];  ai.y = wi[xa * 16 + k1];
    br.x = tBtR[k0 * 64 + ta]; br.y = tBtR[k1 * 64 + ta];
    bi.x = tBtI[k0 * 64 + ta]; bi.y = tBtI[k1 * 64 + ta];
    acc = wmma4(ar, br, acc);
    acc = wmma4(ai, bi, acc);
  }
  float* ob = vnext + (size_t)bo * NP;
#pragma unroll
  for (int vv = 0; vv < 8; ++vv) {
    int x = x0 + vv + 8 * half;
    if (x < XP && ta < TP) ob[x * TP + ta] = acc[vv];   // exec-masked store
  }
}

// ===========================================================================
// Pointwise conv + bias + residual(u1) + optional GELU, weights via LDS:
// vnext[b,o,p] = act( u1 + sum_c vcur[b,c,p] * w[o,c] + bias[o] )
// ===========================================================================
__global__ void k_pconv(const float* __restrict__ vcur, const float* __restrict__ wcv,
                        const float* __restrict__ wb, float* __restrict__ vnext,
                        int apply_gelu, int nwaves) {
  __shared__ float lw[WID * WID];                  // 16 KB of 320 KB/WGP
  for (int i = threadIdx.x; i < WID * WID; i += 256) lw[i] = wcv[i];
  __syncthreads();
  int wid = (int)((blockIdx.x * blockDim.x + threadIdx.x) >> 5);
  if (wid >= nwaves) return;
  int lane = threadIdx.x & 31, half = lane >> 4, l16 = lane & 15;
  int ot = wid & 3;
  int rest = wid >> 2;
  int pt = rest % PT_TILES;
  int b = rest / PT_TILES;
  int p0 = pt * 16;
  int pa = p0 + l16;                               // A row = spatial point
  int oo = ot * 16 + l16;                          // B col = out channel
  const float* vb = vcur + (size_t)b * WID * NP;
  v8f acc = {};
  for (int kb = 0; kb < WID; kb += 4) {            // K = in channel
    int c0 = kb + 2 * half;
    if (kb + 4 < WID)                              // hint next strided A lines
      __builtin_prefetch(&vb[(size_t)(c0 + 4) * NP + pa], 0, 0);
    v2f a, bw;
    a.x = vb[(size_t)c0 * NP + pa];                // unguarded; tail via SLACK
    a.y = vb[(size_t)(c0 + 1) * NP + pa];
    bw.x = lw[oo * WID + c0];                      // w_conv[i][o][c] from LDS
    bw.y = lw[oo * WID + c0 + 1];
    acc = wmma4(a, bw, acc);
  }
  float bias = wb[oo];
  float* ob = vnext + ((size_t)b * WID + oo) * NP;
#pragma unroll
  for (int vv = 0; vv < 8; ++vv) {
    int p = p0 + vv + 8 * half;
    if (p < NP) {
      float val = ob[p] + acc[vv] + bias;          // u1 + u2
      if (apply_gelu) val = gelu_f(val);
      ob[p] = val;
    }
  }
}

// ===========================================================================
// Fused fc1 (64->128, GELU) + fc2 (128->1); fc1 weights via LDS; the 335 MB
// hidden tensor never touches memory. 16-lane xor reduction for the fc2 dot.
// ===========================================================================
__global__ void k_fc1fc2(const float* __restrict__ vcur, const float* __restrict__ f1w,
                         const float* __restrict__ f1b, const float* __restrict__ f2w,
                         const float* __restrict__ f2b, float* __restrict__ out, int nwaves) {
  __shared__ float lw1[WID * 128];                 // 32 KB
  for (int i = threadIdx.x; i < WID * 128; i += 256) lw1[i] = f1w[i];
  __syncthreads();
  int wid = (int)((blockIdx.x * blockDim.x + threadIdx.x) >> 5);
  if (wid >= nwaves) return;
  int lane = threadIdx.x & 31, half = lane >> 4, l16 = lane & 15;
  int pt = wid % 1280;                             // 20480/16 cropped points
  int b = wid / 1280;
  int p0 = pt * 16;
  int pa = p0 + l16;
  int xx = pa / TO, t2 = pa % TO;                  // crop: x<512, t<40
  const float* vb = vcur + (size_t)b * WID * NP;
  v8f acc[8] = {};
  for (int kb = 0; kb < WID; kb += 4) {            // K = c
    int c0 = kb + 2 * half;
    v2f a;
    a.x = vb[(size_t)c0 * NP + xx * TP + t2];
    a.y = vb[(size_t)(c0 + 1) * NP + xx * TP + t2];
#pragma unroll
    for (int nt = 0; nt < 8; ++nt) {
      v2f bw;
      bw.x = lw1[c0 * 128 + nt * 16 + l16];
      bw.y = lw1[(c0 + 1) * 128 + nt * 16 + l16];
      acc[nt] = wmma4(a, bw, acc[nt]);
    }
  }
  float b2 = f2b[0];
#pragma unroll
  for (int vv = 0; vv < 8; ++vv) {
    float s = 0.f;
#pragma unroll
    for (int nt = 0; nt < 8; ++nt) {
      int n = nt * 16 + l16;
      s += gelu_f(acc[nt][vv] + f1b[n]) * f2w[n];
    }
    s += __shfl_xor(s, 1);
    s += __shfl_xor(s, 2);
    s += __shfl_xor(s, 4);
    s += __shfl_xor(s, 8);                         // reduce within 16-lane half
    if (l16 == 0) {
      int m = vv + 8 * half;
      out[(size_t)b * (S_ * TO) + p0 + m] = s + b2;
    }
  }
}

// ===========================================================================
static inline unsigned wblocks(long long nwaves) {
  return (unsigned)((nwaves * 32 + 255) / 256);
}

extern "C" void kernel_launch(void* const* d_in, const int* in_sizes, int n_in,
                              void* d_out, int out_size, void* d_ws, size_t ws_size,
                              hipStream_t stream) {
  (void)in_sizes; (void)n_in; (void)out_size; (void)ws_size;
  const float* u      = (const float*)d_in[0];
  const float* xin    = (const float*)d_in[1];
  const float* tin    = (const float*)d_in[2];
  const float* par    = (const float*)d_in[3];
  const float* fc0w   = (const float*)d_in[4];
  const float* fc0b   = (const float*)d_in[5];
  const float* specwr = (const float*)d_in[6];
  const float* specwi = (const float*)d_in[7];
  const float* wconv  = (const float*)d_in[8];
  const float* wbias  = (const float*)d_in[9];
  const float* fc1w   = (const float*)d_in[10];
  const float* fc1b   = (const float*)d_in[11];
  const float* fc2w   = (const float*)d_in[12];
  const float* fc2b   = (const float*)d_in[13];
  float* out = (float*)d_out;

  // workspace carve-up (floats)
  const size_t NV  = (size_t)B_ * WID * NP + SLACK;   // 209 MB + slack
  const size_t NW1 = (size_t)B_ * WID * XPA * 16;     //  69 MB (X padded to 528)
  const size_t NF  = (size_t)B_ * WID * 256;          //   2 MB
  float* ws   = (float*)d_ws;
  float* vA   = ws;
  float* vB   = vA + NV;
  float* w1r  = vB + NV;                           // also W2 real
  float* w1i  = w1r + NW1;                         // also W2 imag
  float* vftr = w1i + NW1;
  float* vfti = vftr + NF;
  float* corR = vfti + NF;
  float* corI = corR + NF;
  float* tEtR = corI + NF;                         // basis tables (~80 KB)
  float* tEtI = tEtR + TPK * 16;
  float* tCx  = tEtI + TPK * 16;
  float* tSx  = tCx + XPA * 16;
  float* tBtR = tSx + XPA * 16;
  float* tBtI = tBtR + 16 * 64;

  const int T256 = 256;

  // basis tables (XPA*16 = 8448 threads dominate)
  k_tables<<<(XPA * 16 + T256 - 1) / T256, T256, 0, stream>>>(
      tEtR, tEtI, tCx, tSx, tBtR, tBtI);

  // fc0 lift into padded grid
  {
    long long total = (long long)B_ * WID * NP;
    k_fc0<<<(unsigned)((total + T256 - 1) / T256), T256, 0, stream>>>(
        u, xin, tin, par, fc0w, fc0b, vA, total);
  }

  float* vcur = vA;
  float* vnext = vB;
  const long long nw_dftt  = (long long)B_ * WID * XT_TILES;     //  67,584
  const long long nw_dftx  = (long long)B_ * WID;                //   2,048
  const long long nw_mix   = 256 * 2 * 4;                        //   2,048
  const long long nw_idftx = (long long)B_ * WID * XT_TILES;     //  67,584
  const long long nw_idftt = nw_idftx * 4;                       // 270,336
  const long long nw_pconv = (long long)B_ * PT_TILES * 4;       // 204,288

  for (int i = 0; i < 4; ++i) {
    k_dft_t <<<wblocks(nw_dftt),  T256, 0, stream>>>(vcur, tEtR, tEtI,
                 w1r, w1i, (int)nw_dftt);
    k_dft_x <<<wblocks(nw_dftx),  T256, 0, stream>>>(w1r, w1i, tCx, tSx,
                 vftr, vfti, (int)nw_dftx);
    k_mix   <<<wblocks(nw_mix),   T256, 0, stream>>>(vftr, vfti,
                 specwr + (size_t)i * WID * WID * 256,
                 specwi + (size_t)i * WID * WID * 256,
                 corR, corI, (int)nw_mix);
    k_idft_x<<<wblocks(nw_idftx), T256, 0, stream>>>(corR, corI, tCx, tSx,
                 w1r, w1i, (int)nw_idftx);
    k_idft_t<<<wblocks(nw_idftt), T256, 0, stream>>>(w1r, w1i, tBtR, tBtI,
                 vnext, (int)nw_idftt);
    k_pconv <<<wblocks(nw_pconv), T256, 0, stream>>>(vcur,
                 wconv + (size_t)i * WID * WID,
                 wbias + (size_t)i * WID,
                 vnext, (i < 3) ? 1 : 0, (int)nw_pconv);
    float* tmp = vcur; vcur = vnext; vnext = tmp;  // deterministic ping-pong
  }

  const long long nw_fc = (long long)B_ * 1280;    // 40,960 waves
  k_fc1fc2<<<wblocks(nw_fc), T256, 0, stream>>>(vcur, fc1w, fc1b, fc2w, fc2b,
                                                out, (int)nw_fc);
}